// GCNNet_12360915878370
// MI455X (gfx1250) — compile-verified
//
#include <hip/hip_runtime.h>
#include <stdint.h>

#define NN 100000
#define NE 1600000
#define INC 768
#define HID 256
#define OUTC 2

typedef float v2f __attribute__((ext_vector_type(2)));
typedef float v8f __attribute__((ext_vector_type(8)));

// ---------------- graph preprocessing: degree, CSR build ----------------

__global__ __launch_bounds__(256) void init_node_kernel(float* deg, int* cnt, int* fill, int n) {
    int i = blockIdx.x * 256 + threadIdx.x;
    if (i < n) { deg[i] = 1.0f; cnt[i] = 0; fill[i] = 0; }  // deg starts at 1.0 (self loop w=1)
}

__global__ __launch_bounds__(256) void edge_stats_kernel(const int* __restrict__ dst,
                                                         const float* __restrict__ w,
                                                         float* deg, int* cnt, int n) {
    int e = blockIdx.x * 256 + threadIdx.x;
    if (e < n) {
        int d = dst[e];
        atomicAdd(&cnt[d], 1);
        atomicAdd(&deg[d], w[e]);
    }
}

// single-workgroup exclusive scan (100K elements, ~98 chunks of 1024)
__global__ __launch_bounds__(1024) void scan_kernel(const int* __restrict__ cnt,
                                                    int* __restrict__ rowptr, int n) {
    __shared__ int buf[1024];
    __shared__ int carry;
    int tid = threadIdx.x;
    if (tid == 0) carry = 0;
    __syncthreads();
    for (int base = 0; base < n; base += 1024) {
        int v = (base + tid < n) ? cnt[base + tid] : 0;
        buf[tid] = v;
        __syncthreads();
        for (int off = 1; off < 1024; off <<= 1) {
            int t = (tid >= off) ? buf[tid - off] : 0;
            __syncthreads();
            buf[tid] += t;
            __syncthreads();
        }
        if (base + tid < n) rowptr[base + tid] = carry + buf[tid] - v;  // exclusive
        __syncthreads();
        if (tid == 0) carry += buf[1023];
        __syncthreads();
    }
    if (tid == 0) rowptr[n] = carry;
}

__global__ __launch_bounds__(256) void dinv_kernel(const float* __restrict__ deg,
                                                   float* __restrict__ dinv, int n) {
    int i = blockIdx.x * 256 + threadIdx.x;
    if (i < n) {
        float d = deg[i];
        dinv[i] = (d > 0.0f) ? rsqrtf(d) : 0.0f;
    }
}

// fill CSR: col = src, val = dinv[src]*w  (dinv[dst] factored out, applied once per node)
__global__ __launch_bounds__(256) void fill_kernel(const int* __restrict__ src,
                                                   const int* __restrict__ dst,
                                                   const float* __restrict__ w,
                                                   const float* __restrict__ dinv,
                                                   const int* __restrict__ rowptr,
                                                   int* fill, int* __restrict__ col,
                                                   float* __restrict__ val, int n) {
    int e = blockIdx.x * 256 + threadIdx.x;
    if (e < n) {
        int d = dst[e];
        int s = src[e];
        int p = rowptr[d] + atomicAdd(&fill[d], 1);
        col[p] = s;
        val[p] = dinv[s] * w[e];
    }
}

// ---------------- GEMM1: xw1[NN x 256] = x[NN x 768] @ W1[768 x 256] (fp32 WMMA) ------------

__global__ __launch_bounds__(256) void gemm1_wmma(const float* __restrict__ x,
                                                  const float* __restrict__ W1,
                                                  float* __restrict__ xw1) {
    __shared__ float As[64][34];   // pad 34: conflict-free column reads, 8B-aligned pairs
    __shared__ float Bs[32][256];

    int tid  = threadIdx.x;
    int wave = tid >> 5;
    int lane = tid & 31;
    int half = lane >> 4;     // 0: K+0/K+1   1: K+2/K+3
    int lr   = lane & 15;
    int mb   = wave >> 1;     // M sub-tile 0..3
    int nh   = wave & 1;      // N half 0..1 (128 cols each)
    int rowBase = blockIdx.x * 64;

    v8f acc[8] = {};

    for (int k0 = 0; k0 < INC; k0 += 32) {
        // stage A: 64 rows x 32 K  (512 float4, 2 per thread)
        #pragma unroll
        for (int i = 0; i < 2; ++i) {
            int idx = tid + i * 256;
            int r  = idx >> 3;
            int kq = (idx & 7) << 2;
            int gr = rowBase + r;
            float4 v = make_float4(0.f, 0.f, 0.f, 0.f);
            if (gr < NN) v = *(const float4*)(x + (size_t)gr * INC + k0 + kq);
            As[r][kq + 0] = v.x; As[r][kq + 1] = v.y;
            As[r][kq + 2] = v.z; As[r][kq + 3] = v.w;
        }
        // stage B: 32 K x 256 N  (2048 float4, 8 per thread)
        #pragma unroll
        for (int i = 0; i < 8; ++i) {
            int idx = tid + i * 256;
            int kr = idx >> 6;
            int nq = (idx & 63) << 2;
            float4 v = *(const float4*)(W1 + (size_t)(k0 + kr) * HID + nq);
            *(float4*)(&Bs[kr][nq]) = v;
        }
        __syncthreads();

        #pragma unroll
        for (int ks = 0; ks < 32; ks += 4) {
            int ka = ks + 2 * half;
            v2f a;
            a.x = As[mb * 16 + lr][ka];
            a.y = As[mb * 16 + lr][ka + 1];
            #pragma unroll
            for (int nt = 0; nt < 8; ++nt) {
                int cb = nh * 128 + nt * 16 + lr;
                v2f b;
                b.x = Bs[ka][cb];
                b.y = Bs[ka + 1][cb];
                acc[nt] = __builtin_amdgcn_wmma_f32_16x16x4_f32(
                    false, a, false, b, (short)0, acc[nt], false, false);
            }
        }
        __syncthreads();
    }

    // store D: VGPR v holds row (v + 8*half), col = lr within tile
    #pragma unroll
    for (int nt = 0; nt < 8; ++nt) {
        #pragma unroll
        for (int v = 0; v < 8; ++v) {
            int gr = rowBase + mb * 16 + v + 8 * half;
            int gc = nh * 128 + nt * 16 + lr;
            if (gr < NN) xw1[(size_t)gr * HID + gc] = acc[nt][v];
        }
    }
}

// -------- layer-1 aggregation (gather, atomic-free): agg1 = D^-1/2 A D^-1/2 xw1 ----------

__global__ __launch_bounds__(64) void agg1_kernel(const float* __restrict__ xw1,
                                                  const int* __restrict__ rowptr,
                                                  const int* __restrict__ col,
                                                  const float* __restrict__ val,
                                                  const float* __restrict__ dinv,
                                                  float* __restrict__ agg1) {
    int n = blockIdx.x;
    int c4 = threadIdx.x << 2;   // 4 channels per thread (64 threads * 4 = 256)
    float a0 = 0.f, a1 = 0.f, a2 = 0.f, a3 = 0.f;
    int beg = rowptr[n], end = rowptr[n + 1];
    for (int e = beg; e < end; ++e) {
        int s = col[e];
        float v = val[e];                         // dinv[src]*w
        float4 xv = *(const float4*)(xw1 + (size_t)s * HID + c4);
        a0 += v * xv.x; a1 += v * xv.y; a2 += v * xv.z; a3 += v * xv.w;
    }
    float di = dinv[n];
    float4 sv = *(const float4*)(xw1 + (size_t)n * HID + c4);  // self loop
    float4 o;
    o.x = di * a0 + di * di * sv.x;
    o.y = di * a1 + di * di * sv.y;
    o.z = di * a2 + di * di * sv.z;
    o.w = di * a3 + di * di * sv.w;
    *(float4*)(agg1 + (size_t)n * HID + c4) = o;
}

// ---- GEMM2: xw2[NN x 2] = relu(agg1 + b1) @ W2[256 x 2], relu+bias fused into A load ----

__global__ __launch_bounds__(256) void gemm2_wmma(const float* __restrict__ agg1,
                                                  const float* __restrict__ b1,
                                                  const float* __restrict__ W2,
                                                  float* __restrict__ xw2) {
    __shared__ float W2s[HID * OUTC];
    __shared__ float b1s[HID];
    int tid  = threadIdx.x;
    W2s[tid]       = W2[tid];
    W2s[tid + 256] = W2[tid + 256];
    b1s[tid]       = b1[tid];
    __syncthreads();

    int wave = tid >> 5;
    int lane = tid & 31;
    int half = lane >> 4;
    int lr   = lane & 15;
    int rowBase = blockIdx.x * 128 + wave * 16;

    v8f acc = {};
    for (int k0 = 0; k0 < HID; k0 += 4) {
        int ka = k0 + 2 * half;
        int r  = rowBase + lr;
        v2f a;
        float x0 = 0.f, x1 = 0.f;
        if (r < NN) {
            x0 = agg1[(size_t)r * HID + ka]     + b1s[ka];
            x1 = agg1[(size_t)r * HID + ka + 1] + b1s[ka + 1];
        }
        a.x = fmaxf(x0, 0.f);
        a.y = fmaxf(x1, 0.f);
        v2f b;
        b.x = (lr < OUTC) ? W2s[ka * OUTC + lr]       : 0.f;
        b.y = (lr < OUTC) ? W2s[(ka + 1) * OUTC + lr] : 0.f;
        acc = __builtin_amdgcn_wmma_f32_16x16x4_f32(
            false, a, false, b, (short)0, acc, false, false);
    }
    #pragma unroll
    for (int v = 0; v < 8; ++v) {
        int gr = rowBase + v + 8 * half;
        if (gr < NN && lr < OUTC) xw2[(size_t)gr * OUTC + lr] = acc[v];
    }
}

// -------- layer-2 aggregation: out = D^-1/2 A D^-1/2 xw2 + b2  (2 channels/node) --------

__global__ __launch_bounds__(256) void agg2_kernel(const float* __restrict__ xw2,
                                                   const int* __restrict__ rowptr,
                                                   const int* __restrict__ col,
                                                   const float* __restrict__ val,
                                                   const float* __restrict__ dinv,
                                                   const float* __restrict__ b2,
                                                   float* __restrict__ out) {
    int n = blockIdx.x * 256 + threadIdx.x;
    if (n >= NN) return;
    float a0 = 0.f, a1 = 0.f;
    int beg = rowptr[n], end = rowptr[n + 1];
    for (int e = beg; e < end; ++e) {
        int s = col[e];
        float v = val[e];
        a0 += v * xw2[2 * s];
        a1 += v * xw2[2 * s + 1];
    }
    float di = dinv[n];
    out[2 * n]     = di * a0 + di * di * xw2[2 * n]     + b2[0];
    out[2 * n + 1] = di * a1 + di * di * xw2[2 * n + 1] + b2[1];
}

// ---------------------------------- launcher ----------------------------------

extern "C" void kernel_launch(void* const* d_in, const int* in_sizes, int n_in,
                              void* d_out, int out_size, void* d_ws, size_t ws_size,
                              hipStream_t stream) {
    const float* x  = (const float*)d_in[0];
    const int*   ei = (const int*)d_in[1];      // (2, NE): src then dst
    const float* w  = (const float*)d_in[2];
    const float* W1 = (const float*)d_in[3];
    const float* b1 = (const float*)d_in[4];
    const float* W2 = (const float*)d_in[5];
    const float* b2 = (const float*)d_in[6];
    float* out = (float*)d_out;

    const int* src = ei;
    const int* dst = ei + NE;

    // workspace carving (16B aligned regions)
    char* p = (char*)d_ws;
    auto carve = [&](size_t bytes) { char* q = p; p += (bytes + 15) & ~(size_t)15; return q; };
    float* xw1    = (float*)carve((size_t)NN * HID * sizeof(float));   // 102.4 MB
    float* agg1   = (float*)carve((size_t)NN * HID * sizeof(float));   // 102.4 MB
    float* xw2    = (float*)carve((size_t)NN * OUTC * sizeof(float));
    float* deg    = (float*)carve((size_t)NN * sizeof(float));
    float* dinv   = (float*)carve((size_t)NN * sizeof(float));
    float* val    = (float*)carve((size_t)NE * sizeof(float));
    int*   cnt    = (int*)carve((size_t)NN * sizeof(int));
    int*   fill   = (int*)carve((size_t)NN * sizeof(int));
    int*   rowptr = (int*)carve((size_t)(NN + 1) * sizeof(int));
    int*   col    = (int*)carve((size_t)NE * sizeof(int));
    (void)ws_size; (void)n_in; (void)in_sizes; (void)out_size;

    int gN = (NN + 255) / 256;
    int gE = (NE + 255) / 256;

    init_node_kernel<<<gN, 256, 0, stream>>>(deg, cnt, fill, NN);
    edge_stats_kernel<<<gE, 256, 0, stream>>>(dst, w, deg, cnt, NE);
    scan_kernel<<<1, 1024, 0, stream>>>(cnt, rowptr, NN);
    dinv_kernel<<<gN, 256, 0, stream>>>(deg, dinv, NN);
    fill_kernel<<<gE, 256, 0, stream>>>(src, dst, w, dinv, rowptr, fill, col, val, NE);

    gemm1_wmma<<<(NN + 63) / 64, 256, 0, stream>>>(x, W1, xw1);
    agg1_kernel<<<NN, 64, 0, stream>>>(xw1, rowptr, col, val, dinv, agg1);
    gemm2_wmma<<<(NN + 127) / 128, 256, 0, stream>>>(agg1, b1, W2, xw2);
    agg2_kernel<<<gN, 256, 0, stream>>>(xw2, rowptr, col, val, dinv, b2, out);
}